// GCN_11776800326010
// MI455X (gfx1250) — compile-verified
//
#include <hip/hip_runtime.h>
#include <hip/hip_bf16.h>
#include <math.h>

typedef __attribute__((ext_vector_type(2))) float v2f;
typedef __attribute__((ext_vector_type(8))) float v8f;

#define NFEAT 512
#define HID   16
#define NCLS  40
#define NPAD  48

// ---------------- degree / normalization ----------------

__global__ void deg_init_kernel(float* deg, int n) {
    int i = blockIdx.x * blockDim.x + threadIdx.x;
    if (i < n) deg[i] = 1.0f;   // self-loop
}

__global__ void deg_count_kernel(const int* __restrict__ dst, float* deg, int nE) {
    int e = blockIdx.x * blockDim.x + threadIdx.x;
    if (e < nE) atomicAdd(&deg[dst[e]], 1.0f);
}

__global__ void deg_rsqrt_kernel(float* deg, int n) {
    int i = blockIdx.x * blockDim.x + threadIdx.x;
    if (i < n) {
        float d = deg[i];
        deg[i] = (d > 0.0f) ? rsqrtf(d) : 0.0f;   // becomes dinv
    }
}

// ---------------- GEMM1: h1 = x @ W1  (fp32 WMMA 16x16x4) ----------------
// One wave computes a 16(M) x 16(N) tile; K = 512 -> 128 WMMA steps.

__global__ __launch_bounds__(256) void gemm1_wmma_kernel(
        const float* __restrict__ x, const float* __restrict__ W1,
        float* __restrict__ h1, int n) {
    __shared__ float sW[NFEAT * HID];   // 32 KB
    for (int i = threadIdx.x; i < NFEAT * HID; i += 256) sW[i] = W1[i];
    __syncthreads();

    int wave = threadIdx.x >> 5;
    int lane = threadIdx.x & 31;
    long mtile = (long)blockIdx.x * 8 + wave;
    long row0  = mtile * 16;
    if (row0 >= n) return;              // uniform per wave

    int m  = lane & 15;                 // A: row / B,D: column / D: N
    int kh = lane >> 4;                 // 0 or 1 -> selects k-pair
    long row = row0 + m;
    if (row >= n) row = n - 1;          // clamp; garbage rows never stored
    const float* arow = x + row * NFEAT + 2 * kh;

    v8f acc = (v8f)0.0f;
    for (int ks = 0; ks < NFEAT; ks += 4) {
        v2f a, b;
        a.x = arow[ks + 0];
        a.y = arow[ks + 1];
        int kb = ks + 2 * kh;
        b.x = sW[(kb + 0) * HID + m];
        b.y = sW[(kb + 1) * HID + m];
        acc = __builtin_amdgcn_wmma_f32_16x16x4_f32(
                false, a, false, b, (short)0, acc, false, false);
    }
    // D layout: lane l holds N = l%16, M = v + 8*(l/16)
    for (int v = 0; v < 8; ++v) {
        long mr = row0 + v + 8 * kh;
        if (mr < n) h1[mr * HID + m] = acc[v];
    }
}

// ---------------- aggregation layer 1 ----------------

__global__ void selfloop1_kernel(const float* __restrict__ h1,
                                 const float* __restrict__ dinv,
                                 float* __restrict__ agg, int n) {
    long idx = (long)blockIdx.x * blockDim.x + threadIdx.x;
    if (idx >= (long)n * HID) return;
    int i = (int)(idx >> 4);
    float di = dinv[i];
    agg[idx] = h1[idx] * di * di;
}

__global__ void scatter1_kernel(const int* __restrict__ src,
                                const int* __restrict__ dst,
                                const float* __restrict__ h1,
                                const float* __restrict__ dinv,
                                float* __restrict__ agg, int nE) {
    long idx = (long)blockIdx.x * blockDim.x + threadIdx.x;
    if (idx >= (long)nE * HID) return;
    int e = (int)(idx >> 4);
    int j = (int)(idx & 15);
    int s = src[e], d = dst[e];
    float norm = dinv[s] * dinv[d];
    atomicAdd(&agg[(long)d * HID + j], h1[(long)s * HID + j] * norm);
}

__global__ void bias_relu_kernel(float* __restrict__ agg,
                                 const float* __restrict__ b1, int n) {
    long idx = (long)blockIdx.x * blockDim.x + threadIdx.x;
    if (idx >= (long)n * HID) return;
    int j = (int)(idx & 15);
    float v = agg[idx] + b1[j];
    agg[idx] = v > 0.0f ? v : 0.0f;     // in-place -> h1r
}

// ---------------- GEMM2: h2 = h1r @ W2  (K=16, N=40 padded to 48) ----------------

__global__ __launch_bounds__(256) void gemm2_wmma_kernel(
        const float* __restrict__ h1r, const float* __restrict__ W2,
        float* __restrict__ h2, int n) {
    __shared__ float sW[HID * NPAD];    // 16 x 48, zero-padded
    for (int i = threadIdx.x; i < HID * NPAD; i += 256) {
        int k = i / NPAD, c = i % NPAD;
        sW[i] = (c < NCLS) ? W2[k * NCLS + c] : 0.0f;
    }
    __syncthreads();

    int wave = threadIdx.x >> 5;
    int lane = threadIdx.x & 31;
    long mtile = (long)blockIdx.x * 8 + wave;
    long row0  = mtile * 16;
    if (row0 >= n) return;

    int m  = lane & 15;
    int kh = lane >> 4;
    long row = row0 + m;
    if (row >= n) row = n - 1;

    v2f afrag[4];
    for (int ks = 0; ks < 4; ++ks) {
        afrag[ks].x = h1r[row * HID + ks * 4 + 2 * kh + 0];
        afrag[ks].y = h1r[row * HID + ks * 4 + 2 * kh + 1];
    }

    for (int nt = 0; nt < 3; ++nt) {
        v8f acc = (v8f)0.0f;
        for (int ks = 0; ks < 4; ++ks) {
            v2f b;
            int kb = ks * 4 + 2 * kh;
            b.x = sW[(kb + 0) * NPAD + nt * 16 + m];
            b.y = sW[(kb + 1) * NPAD + nt * 16 + m];
            acc = __builtin_amdgcn_wmma_f32_16x16x4_f32(
                    false, afrag[ks], false, b, (short)0, acc, false, false);
        }
        int col = nt * 16 + m;
        if (col < NCLS) {
            for (int v = 0; v < 8; ++v) {
                long mr = row0 + v + 8 * kh;
                if (mr < n) h2[mr * NCLS + col] = acc[v];
            }
        }
    }
}

// ---------------- aggregation layer 2 ----------------

__global__ void selfloop2_kernel(const float* __restrict__ h2,
                                 const float* __restrict__ dinv,
                                 float* __restrict__ agg, int n) {
    long idx = (long)blockIdx.x * blockDim.x + threadIdx.x;
    if (idx >= (long)n * NCLS) return;
    int i = (int)(idx / NCLS);
    float di = dinv[i];
    agg[idx] = h2[idx] * di * di;
}

__global__ void scatter2_kernel(const int* __restrict__ src,
                                const int* __restrict__ dst,
                                const float* __restrict__ h2,
                                const float* __restrict__ dinv,
                                float* __restrict__ agg, int nE) {
    long idx = (long)blockIdx.x * blockDim.x + threadIdx.x;
    if (idx >= (long)nE * NCLS) return;
    int e = (int)(idx / NCLS);
    int j = (int)(idx - (long)e * NCLS);
    int s = src[e], d = dst[e];
    float norm = dinv[s] * dinv[d];
    atomicAdd(&agg[(long)d * NCLS + j], h2[(long)s * NCLS + j] * norm);
}

// ---------------- bias + log_softmax ----------------

__global__ void logsoftmax_kernel(const float* __restrict__ agg,
                                  const float* __restrict__ b2,
                                  float* __restrict__ out, int n) {
    int i = blockIdx.x * blockDim.x + threadIdx.x;
    if (i >= n) return;
    float z[NCLS];
    float mx = -INFINITY;
    const float* row = agg + (long)i * NCLS;
    for (int j = 0; j < NCLS; ++j) {
        z[j] = row[j] + b2[j];
        mx = fmaxf(mx, z[j]);
    }
    float s = 0.0f;
    for (int j = 0; j < NCLS; ++j) s += expf(z[j] - mx);
    float lse = mx + logf(s);
    float* orow = out + (long)i * NCLS;
    for (int j = 0; j < NCLS; ++j) orow[j] = z[j] - lse;
}

// ---------------- launch ----------------

static inline size_t align_up(size_t v, size_t a) { return (v + a - 1) & ~(a - 1); }

extern "C" void kernel_launch(void* const* d_in, const int* in_sizes, int n_in,
                              void* d_out, int out_size, void* d_ws, size_t ws_size,
                              hipStream_t stream) {
    const float* x   = (const float*)d_in[0];
    const int*   ei  = (const int*)d_in[1];
    const float* W1  = (const float*)d_in[2];
    const float* b1  = (const float*)d_in[3];
    const float* W2  = (const float*)d_in[4];
    const float* b2  = (const float*)d_in[5];
    float* out = (float*)d_out;

    const int n  = in_sizes[0] / NFEAT;     // 100000
    const int nE = in_sizes[1] / 2;         // 3200000
    const int* src = ei;
    const int* dst = ei + nE;

    // workspace partition (floats)
    char* ws = (char*)d_ws;
    size_t off = 0;
    float* dinv = (float*)(ws + off); off = align_up(off + (size_t)n * 4, 256);
    float* h1   = (float*)(ws + off); off = align_up(off + (size_t)n * HID * 4, 256);
    float* agg1 = (float*)(ws + off); off = align_up(off + (size_t)n * HID * 4, 256);
    float* h2   = (float*)(ws + off); off = align_up(off + (size_t)n * NCLS * 4, 256);
    float* agg2 = (float*)(ws + off); off = align_up(off + (size_t)n * NCLS * 4, 256);
    (void)ws_size;

    const int B = 256;
    // normalization
    deg_init_kernel<<<(n + B - 1) / B, B, 0, stream>>>(dinv, n);
    deg_count_kernel<<<(nE + B - 1) / B, B, 0, stream>>>(dst, dinv, nE);
    deg_rsqrt_kernel<<<(n + B - 1) / B, B, 0, stream>>>(dinv, n);

    // layer 1
    int g1 = (int)(((long)n + 127) / 128);  // 8 waves x 16 rows per block
    gemm1_wmma_kernel<<<g1, B, 0, stream>>>(x, W1, h1, n);
    long nh1 = (long)n * HID;
    selfloop1_kernel<<<(int)((nh1 + B - 1) / B), B, 0, stream>>>(h1, dinv, agg1, n);
    long ne1 = (long)nE * HID;
    scatter1_kernel<<<(int)((ne1 + B - 1) / B), B, 0, stream>>>(src, dst, h1, dinv, agg1, nE);
    bias_relu_kernel<<<(int)((nh1 + B - 1) / B), B, 0, stream>>>(agg1, b1, n);

    // layer 2
    gemm2_wmma_kernel<<<g1, B, 0, stream>>>(agg1, W2, h2, n);
    long nh2 = (long)n * NCLS;
    selfloop2_kernel<<<(int)((nh2 + B - 1) / B), B, 0, stream>>>(h2, dinv, agg2, n);
    long ne2 = (long)nE * NCLS;
    scatter2_kernel<<<(int)((ne2 + B - 1) / B), B, 0, stream>>>(src, dst, h2, dinv, agg2, nE);

    // output
    logsoftmax_kernel<<<(n + B - 1) / B, B, 0, stream>>>(agg2, b2, out, n);
}